// ProductManifoldGPHead_6390911336719
// MI455X (gfx1250) — compile-verified
//
#include <hip/hip_runtime.h>
#include <hip/hip_bf16.h>

// ---------------------------------------------------------------------------
// ProductManifoldGPHead for MI455X (gfx1250, wave32).
// mean = Kxu @ (Ku^{-1} f_u), Kxu = Kh(poincare) * Kz(rbf).
// fp32 WMMA 16x16x4 for the z-Gram cross term and the final [BT,256]x[256,16]
// GEMM. u_z and W are pre-swizzled into WMMA B-fragment order so all hot-loop
// B loads are single coalesced global_load_b64; B fragments are preloaded in
// bulk (clause) before the WMMA stream; dual accumulators break XDL RAW chains.
// ---------------------------------------------------------------------------

typedef __attribute__((ext_vector_type(2))) float v2f;
typedef __attribute__((ext_vector_type(8))) float v8f;

#define EPS_F       1e-6f
#define BALL_EDGE_F (1.0f - 1e-5f)
#define MAX_SQN_F   (1.0f - 1e-6f)
#define JITTER_F    1e-5f

constexpr int Zc  = 64;        // z feature dim
constexpr int Mc  = 256;       // inducing points
constexpr int Ac  = 16;        // action dim
constexpr int Wd  = Mc + Ac;   // augmented width (272)
constexpr int KCc = Zc / 4;    // 16 WMMA K-chunks for the Gram tile
constexpr int MTc = Mc / 16;   // 16 inducing tiles

// ---- kernel 0: per-row stats for X: projected h, clamped |h|^2, |z|^2 -----
__global__ void prep_rows(const float* __restrict__ h, const float* __restrict__ z,
                          float4* __restrict__ rowdata, int BT) {
    int r = blockIdx.x * blockDim.x + threadIdx.x;
    if (r >= BT) return;
    float h0 = h[2 * r], h1 = h[2 * r + 1];
    float rr = sqrtf(h0 * h0 + h1 * h1);
    if (rr >= BALL_EDGE_F) {
        float s = BALL_EDGE_F / fmaxf(rr, 1.0f);
        h0 *= s; h1 *= s;
    }
    float x2 = fminf(h0 * h0 + h1 * h1, MAX_SQN_F);
    const float4* zr = (const float4*)(z + (size_t)r * Zc);
    float zz = 0.0f;
#pragma unroll
    for (int k = 0; k < Zc / 4; ++k) {
        float4 q = zr[k];
        zz += q.x * q.x + q.y * q.y + q.z * q.z + q.w * q.w;
    }
    rowdata[r] = make_float4(h0, h1, x2, zz);
}

// ---- kernel 1: same stats for inducing points ------------------------------
__global__ void prep_u(const float* __restrict__ uh, const float* __restrict__ uz,
                       float4* __restrict__ udata) {
    int m = blockIdx.x * blockDim.x + threadIdx.x;
    if (m >= Mc) return;
    float h0 = uh[2 * m], h1 = uh[2 * m + 1];
    float rr = sqrtf(h0 * h0 + h1 * h1);
    if (rr >= BALL_EDGE_F) {
        float s = BALL_EDGE_F / fmaxf(rr, 1.0f);
        h0 *= s; h1 *= s;
    }
    float x2 = fminf(h0 * h0 + h1 * h1, MAX_SQN_F);
    const float4* zr = (const float4*)(uz + (size_t)m * Zc);
    float zz = 0.0f;
#pragma unroll
    for (int k = 0; k < Zc / 4; ++k) {
        float4 q = zr[k];
        zz += q.x * q.x + q.y * q.y + q.z * q.z + q.w * q.w;
    }
    udata[m] = make_float4(h0, h1, x2, zz);
}

__device__ __forceinline__ float poincare_kh(float4 a, float4 b, float inv2h) {
    float d0 = a.x - b.x, d1 = a.y - b.y;
    float diff2 = d0 * d0 + d1 * d1;
    float den = (1.0f - a.z) * (1.0f - b.z) + EPS_F;
    float arg = 1.0f + 2.0f * diff2 / den;
    float s = sqrtf(fmaxf(arg * arg - 1.0f, EPS_F));
    float d = __logf(arg + s);
    return __expf(-d * d * inv2h);
}

// ---- kernel 2: augmented [Ku | f_u], width 272 ----------------------------
__global__ void build_ku(const float4* __restrict__ udata, const float* __restrict__ uz,
                         const float* __restrict__ fu,
                         const float* __restrict__ lsh, const float* __restrict__ lsz,
                         float* __restrict__ aug) {
    int i = blockIdx.x;
    int j = threadIdx.x;
    float inv2h = 0.5f * __expf(-2.0f * lsh[0]);
    float inv2z = 0.5f * __expf(-2.0f * lsz[0]);
    float4 ui = udata[i], uj = udata[j];
    const float4* zi = (const float4*)(uz + (size_t)i * Zc);
    const float4* zj = (const float4*)(uz + (size_t)j * Zc);
    float d2 = 0.0f;
#pragma unroll
    for (int k = 0; k < Zc / 4; ++k) {
        float4 a = zi[k], b = zj[k];
        float e0 = a.x - b.x, e1 = a.y - b.y, e2 = a.z - b.z, e3 = a.w - b.w;
        d2 += e0 * e0 + e1 * e1 + e2 * e2 + e3 * e3;
    }
    float val = poincare_kh(ui, uj, inv2h) * __expf(-d2 * inv2z);
    if (i == j) val += JITTER_F;
    aug[(size_t)i * Wd + j] = val;
    if (j < Ac) aug[(size_t)i * Wd + Mc + j] = fu[i * Ac + j];
}

// ---- kernel 3: one-block Gauss-Jordan, thread = row -----------------------
__global__ void gj_solve(float* __restrict__ aug) {
    __shared__ float piv[Wd];
    int r = threadIdx.x;
    for (int k = 0; k < Mc; ++k) {
        for (int j = threadIdx.x; j < Wd; j += Mc) piv[j] = aug[(size_t)k * Wd + j];
        __syncthreads();
        float pivval = piv[k];
        float factor = aug[(size_t)r * Wd + k] / pivval;
        if (r != k) {
            for (int j = k; j < Wd; ++j) aug[(size_t)r * Wd + j] -= factor * piv[j];
        }
        __syncthreads();
        if (r == k) {
            float inv = 1.0f / pivval;
            for (int j = k; j < Wd; ++j) aug[(size_t)k * Wd + j] = piv[j] * inv;
        }
        __syncthreads();
    }
}

// ---- kernel 3b: W = Ku^{-1} f_u, pre-swizzled to WMMA B-fragment order ----
// wfrag[(mt*4 + kc2)*32 + lane] = { W[mt*16 + k0][nl], W[mt*16 + k0 + 1][nl] }
// with k0 = 4*kc2 + 2*(lane>>4), nl = lane&15.
__global__ void extract_w(const float* __restrict__ aug, v2f* __restrict__ wfrag) {
    int tid  = blockIdx.x * blockDim.x + threadIdx.x;   // 0 .. MTc*4*32-1
    int lane = tid & 31;
    int kc2  = (tid >> 5) & 3;
    int mt   = tid >> 7;
    int hi = lane >> 4, nl = lane & 15;
    int k0 = 4 * kc2 + 2 * hi;
    v2f w;
    w.x = aug[(size_t)(mt * 16 + k0) * Wd + Mc + nl];
    w.y = aug[(size_t)(mt * 16 + k0 + 1) * Wd + Mc + nl];
    wfrag[tid] = w;
}

// ---- kernel 3c: u_z pre-swizzled to WMMA B-fragment order -----------------
// bzfrag[(mt*KCc + kc)*32 + lane] = { uz[(mt*16+nl)*Zc + 4*kc + 2*hi + {0,1}] }
__global__ void swizzle_uz(const float* __restrict__ uz, v2f* __restrict__ bzfrag) {
    int tid  = blockIdx.x * blockDim.x + threadIdx.x;   // 0 .. MTc*KCc*32-1
    int lane = tid & 31;
    int kc   = (tid >> 5) & (KCc - 1);
    int mt   = tid >> 9;
    int hi = lane >> 4, nl = lane & 15;
    v2f b;
    b.x = uz[(size_t)(mt * 16 + nl) * Zc + 4 * kc + 2 * hi];
    b.y = uz[(size_t)(mt * 16 + nl) * Zc + 4 * kc + 2 * hi + 1];
    bzfrag[tid] = b;
}

// ---- kernel 4: fused Kxu + output GEMM, one 16-row tile per wave32 --------
__global__ __launch_bounds__(256) void gp_head(
    const float* __restrict__ z,
    const v2f* __restrict__ bzfrag, const v2f* __restrict__ wfrag,
    const float4* __restrict__ rowdata, const float4* __restrict__ udata,
    const float* __restrict__ lsh, const float* __restrict__ lsz,
    float* __restrict__ out) {
    const int lane = threadIdx.x & 31;
    const int wave = threadIdx.x >> 5;
    const int tile = blockIdx.x * (blockDim.x >> 5) + wave;
    const int hi   = lane >> 4;     // half-wave select per ISA fragment layout
    const int nl   = lane & 15;
    const int row0 = tile * 16;

    const float inv2h = 0.5f * __expf(-2.0f * lsh[0]);
    const float inv2z = 0.5f * __expf(-2.0f * lsz[0]);

    // per-wave 16x16 Kxu tile; row stride 18 floats (even -> 8B-aligned pairs,
    // 72B lane stride is bank-conflict-free on 64 banks)
    __shared__ float kx[8][16][18];

    // Per-lane row stats for output rows M = v + 8*hi (C/D layout rows).
    float4 rd[8];
#pragma unroll
    for (int v = 0; v < 8; ++v) rd[v] = rowdata[row0 + v + 8 * hi];

    // A fragments for the z Gram GEMM, loaded once:
    // lane holds row M = lane&15, K = 4*kc + 2*hi + {0,1}  (contiguous pair).
    const v2f* zrow = (const v2f*)(z + (size_t)(row0 + nl) * Zc + 2 * hi);
    v2f az[KCc];
#pragma unroll
    for (int kc = 0; kc < KCc; ++kc) az[kc] = zrow[2 * kc];

    v8f outA = {0.f, 0.f, 0.f, 0.f, 0.f, 0.f, 0.f, 0.f};
    v8f outB = {0.f, 0.f, 0.f, 0.f, 0.f, 0.f, 0.f, 0.f};

    for (int mt = 0; mt < MTc; ++mt) {
        // ---- stage 1a: bulk-load all 16 B fragments (coalesced b64 clause) -
        const v2f* bz = bzfrag + (size_t)mt * (KCc * 32) + lane;
        v2f bf[KCc];
#pragma unroll
        for (int kc = 0; kc < KCc; ++kc) bf[kc] = bz[kc * 32];

        // ---- stage 1b: G = Xz_tile @ uz_tile^T, dual accumulators ---------
        v8f g0 = {0.f, 0.f, 0.f, 0.f, 0.f, 0.f, 0.f, 0.f};
        v8f g1 = {0.f, 0.f, 0.f, 0.f, 0.f, 0.f, 0.f, 0.f};
#pragma unroll
        for (int kc = 0; kc < KCc; kc += 2) {
            g0 = __builtin_amdgcn_wmma_f32_16x16x4_f32(
                     false, az[kc], false, bf[kc], (short)0, g0, false, false);
            g1 = __builtin_amdgcn_wmma_f32_16x16x4_f32(
                     false, az[kc + 1], false, bf[kc + 1], (short)0, g1, false, false);
        }

        // ---- stage 2: elementwise product kernel (co-executes with XDL) ---
        float4 ud = udata[mt * 16 + nl];
        float kxv[8];
#pragma unroll
        for (int v = 0; v < 8; ++v) {
            float dot = g0[v] + g1[v];
            float d2z = fmaxf(rd[v].w + ud.w - 2.0f * dot, 0.0f);
            float kz  = __expf(-d2z * inv2z);
            float d0 = rd[v].x - ud.x, d1 = rd[v].y - ud.y;
            float diff2 = d0 * d0 + d1 * d1;
            float den = (1.0f - rd[v].z) * (1.0f - ud.z) + EPS_F;
            float arg = 1.0f + 2.0f * diff2 / den;
            float s = sqrtf(fmaxf(arg * arg - 1.0f, EPS_F));
            float dP = __logf(arg + s);
            kxv[v] = __expf(-dP * dP * inv2h) * kz;
        }

        // ---- stage 3: C/D layout -> A layout through LDS ------------------
        __syncthreads();                        // protect previous iter reads
#pragma unroll
        for (int v = 0; v < 8; ++v) kx[wave][v + 8 * hi][nl] = kxv[v];
        __syncthreads();

        // ---- stage 4: out += Kxu_tile(16x16) @ W_tile(16x16) --------------
        const v2f* wfp = wfrag + (size_t)mt * (4 * 32) + lane;
        v2f wf[4];
#pragma unroll
        for (int kc2 = 0; kc2 < 4; ++kc2) wf[kc2] = wfp[kc2 * 32];
#pragma unroll
        for (int kc2 = 0; kc2 < 4; ++kc2) {
            int k0 = 4 * kc2 + 2 * hi;
            v2f a2 = *(const v2f*)&kx[wave][nl][k0];   // ds_load_b64
            if (kc2 & 1)
                outB = __builtin_amdgcn_wmma_f32_16x16x4_f32(
                           false, a2, false, wf[kc2], (short)0, outB, false, false);
            else
                outA = __builtin_amdgcn_wmma_f32_16x16x4_f32(
                           false, a2, false, wf[kc2], (short)0, outA, false, false);
        }
    }

    // D layout: VGPR v -> row v + 8*hi, col nl.
#pragma unroll
    for (int v = 0; v < 8; ++v)
        out[(size_t)(row0 + v + 8 * hi) * Ac + nl] = outA[v] + outB[v];
}

// ---------------------------------------------------------------------------
extern "C" void kernel_launch(void* const* d_in, const int* in_sizes, int n_in,
                              void* d_out, int out_size, void* d_ws, size_t ws_size,
                              hipStream_t stream) {
    const float* h   = (const float*)d_in[0];   // [B,T,2]
    const float* z   = (const float*)d_in[1];   // [B,T,64]
    const float* uh  = (const float*)d_in[2];   // [256,2]
    const float* uz  = (const float*)d_in[3];   // [256,64]
    const float* lsh = (const float*)d_in[4];   // scalar
    const float* lsz = (const float*)d_in[5];   // scalar
    const float* fu  = (const float*)d_in[6];   // [256,16]

    const int BT = in_sizes[0] / 2;             // 8192

    // ws: rowdata[BT] f4 | udata[256] f4 | aug[256][272] | wfrag[2048] v2f | bzfrag[8192] v2f
    float4* rowdata = (float4*)d_ws;
    float4* udata   = rowdata + BT;
    float*  aug     = (float*)(udata + Mc);
    v2f*    wfrag   = (v2f*)(aug + (size_t)Mc * Wd);
    v2f*    bzfrag  = wfrag + MTc * 4 * 32;

    prep_rows<<<(BT + 255) / 256, 256, 0, stream>>>(h, z, rowdata, BT);
    prep_u<<<1, Mc, 0, stream>>>(uh, uz, udata);
    build_ku<<<Mc, Mc, 0, stream>>>(udata, uz, fu, lsh, lsz, aug);
    gj_solve<<<1, Mc, 0, stream>>>(aug);
    extract_w<<<(MTc * 4 * 32) / 256, 256, 0, stream>>>(aug, wfrag);
    swizzle_uz<<<(MTc * KCc * 32) / 256, 256, 0, stream>>>(uz, bzfrag);

    const int wavesPerBlock = 8;                // 256 threads
    const int nTiles = BT / 16;                 // 512 row tiles
    gp_head<<<nTiles / wavesPerBlock, 32 * wavesPerBlock, 0, stream>>>(
        z, bzfrag, wfrag, rowdata, udata, lsh, lsz, (float*)d_out);
}